// Atom_Query_37306085933337
// MI455X (gfx1250) — compile-verified
//
#include <hip/hip_runtime.h>
#include <hip/hip_bf16.h>
#include <math.h>

// CDNA5 (gfx1250) wave32 WMMA f32 path: V_WMMA_F32_16X16X4_F32
typedef float v2f __attribute__((ext_vector_type(2)));
typedef float v8f __attribute__((ext_vector_type(8)));

#define BDIM 4
#define CDIM 128
#define NDIM 4096
#define MDIM 8192
#define KNN  16
#define EPSBN 1e-5f
#define XSTRIDE 17   // LDS row pad (stride-17 kills 4-way bank conflicts)

// ---------------------------------------------------------------------------
// Kernel 1: transpose atom_feats (B,C,N) -> featsT (B,N,C) via 32x32 LDS tile
// so each KNN gather reads a contiguous 512B feature row.
// ---------------------------------------------------------------------------
__global__ void feats_transpose_kernel(const float* __restrict__ feats,
                                       float* __restrict__ featsT) {
  __shared__ float tile[32][33];
  const int b  = blockIdx.z;
  const int n0 = blockIdx.x * 32;
  const int c0 = blockIdx.y * 32;
  const int tx = threadIdx.x;   // 0..31
  const int ty = threadIdx.y;   // 0..7
  const float* src = feats  + (size_t)b * CDIM * NDIM;
  float*       dst = featsT + (size_t)b * NDIM * CDIM;
#pragma unroll
  for (int i = 0; i < 4; ++i)
    tile[ty + 8 * i][tx] = src[(size_t)(c0 + ty + 8 * i) * NDIM + (n0 + tx)];
  __syncthreads();
#pragma unroll
  for (int i = 0; i < 4; ++i)
    dst[(size_t)(n0 + ty + 8 * i) * CDIM + (c0 + tx)] = tile[tx][ty + 8 * i];
}

// ---------------------------------------------------------------------------
// Kernel 2: transpose W3 (128 x 256) -> W3T (256 x 128) for coalesced conv3.
// ---------------------------------------------------------------------------
__global__ void w3_transpose_kernel(const float* __restrict__ w3,
                                    float* __restrict__ w3t) {
  int t = blockIdx.x * 256 + threadIdx.x;   // 0..32767
  int c = t >> 8;                           // W3 row  (0..127)
  int i = t & 255;                          // W3 col  (0..255)
  w3t[i * CDIM + c] = w3[c * 256 + i];
}

// ---------------------------------------------------------------------------
// Kernel 3: KNN top-16. One thread per surf point; atoms streamed via LDS.
// Guarded, fully-unrolled register insertion keeps top-16 sorted ascending.
// ---------------------------------------------------------------------------
__global__ void __launch_bounds__(256)
knn_kernel(const float* __restrict__ atom_xyz,   // (B,N,3)
           const float* __restrict__ surf_xyz,   // (B,M,3)
           int* __restrict__ out_idx,            // (B*M,K)
           float* __restrict__ out_dist) {       // (B*M,K)
  __shared__ float axs[256 * 3];
  const int t = blockIdx.x * 256 + threadIdx.x;  // block spans one b (256|8192)
  const int b = t / MDIM;
  const float* sp = surf_xyz + (size_t)t * 3;
  const float sx = sp[0], sy = sp[1], sz = sp[2];

  float bd[KNN]; int bi[KNN];
#pragma unroll
  for (int k = 0; k < KNN; ++k) { bd[k] = 3.4e38f; bi[k] = 0; }

  const float* ab = atom_xyz + (size_t)b * NDIM * 3;
  for (int tile = 0; tile < NDIM; tile += 256) {
    __syncthreads();
#pragma unroll
    for (int j = 0; j < 3; ++j)
      axs[threadIdx.x + 256 * j] = ab[(size_t)tile * 3 + threadIdx.x + 256 * j];
    __syncthreads();
    for (int j = 0; j < 256; ++j) {
      float dx = axs[3 * j + 0] - sx;
      float dy = axs[3 * j + 1] - sy;
      float dz = axs[3 * j + 2] - sz;
      float cd = dx * dx + dy * dy + dz * dz;
      if (cd < bd[KNN - 1]) {          // rarely taken after warmup
        int ci = tile + j;
#pragma unroll
        for (int k = 0; k < KNN; ++k) {  // bubble-insert, static indices only
          if (cd < bd[k]) {
            float td = bd[k]; bd[k] = cd; cd = td;
            int   ti = bi[k]; bi[k] = ci; ci = ti;
          }
        }
      }
    }
  }
#pragma unroll
  for (int k = 0; k < KNN; ++k) {
    out_idx [(size_t)t * KNN + k] = bi[k];
    out_dist[(size_t)t * KNN + k] = sqrtf(bd[k]);
  }
}

// ---------------------------------------------------------------------------
// Kernel 4: fused gather + conv1 + conv2 (+K-sums) + conv3, all BN/bias/lrelu
// folded. 8 waves/WG; wave w owns C-rows [16w,16w+16) of W1/W2 in REGISTERS.
// Per point: X(132x16) in LDS, 33+32 steps of v_wmma_f32_16x16x4_f32.
// ---------------------------------------------------------------------------
__global__ void __launch_bounds__(256, 1)
fused_kernel(const float* __restrict__ featsT,   // (B,N,C)
             const int*   __restrict__ knn_idx,  // (B*M,K)
             const float* __restrict__ knn_dist, // (B*M,K)
             const float* __restrict__ w1,  const float* __restrict__ b1,
             const float* __restrict__ g1,  const float* __restrict__ be1,
             const float* __restrict__ mu1, const float* __restrict__ va1,
             const float* __restrict__ w2,  const float* __restrict__ b2,
             const float* __restrict__ g2,  const float* __restrict__ be2,
             const float* __restrict__ mu2, const float* __restrict__ va2,
             const float* __restrict__ w3t, const float* __restrict__ b3,
             const float* __restrict__ g3,  const float* __restrict__ be3,
             const float* __restrict__ mu3, const float* __restrict__ va3,
             float* __restrict__ out) {          // (B,C,M)
  __shared__ float Xs[132 * XSTRIDE];   // gathered [feats;dist;0-pad] 132x16
  __shared__ float Ys[128 * XSTRIDE];   // conv1 activation 128x16
  __shared__ float Ss[256];             // [s1;s2]
  __shared__ float sc1s[128], sh1s[128], sc2s[128], sh2s[128], sc3s[128], sh3s[128];
  __shared__ int   nbr[16];

  const int tid  = threadIdx.x;
  const int lane = tid & 31;
  const int wave = tid >> 5;       // 0..7
  const int hh   = lane >> 4;      // half-wave: 0/1
  const int col  = lane & 15;      // N (neighbor) index / A-row index
  const int slab = wave * 16;      // first C-row of this wave's output tile

  // Fold BN+bias once per workgroup: y = x*sc + sh, sh = beta - mu*sc + b*sc
  if (tid < 128) {
    int c = tid;
    float s1v = g1[c] * rsqrtf(va1[c] + EPSBN);
    sc1s[c] = s1v; sh1s[c] = be1[c] - mu1[c] * s1v + b1[c] * s1v;
    float s2v = g2[c] * rsqrtf(va2[c] + EPSBN);
    sc2s[c] = s2v; sh2s[c] = be2[c] - mu2[c] * s2v + b2[c] * s2v;
    float s3v = g3[c] * rsqrtf(va3[c] + EPSBN);
    sc3s[c] = s3v; sh3s[c] = be3[c] - mu3[c] * s3v + b3[c] * s3v;
  }
  // Constant zero pad rows 129..131 of X (K padded 129 -> 132)
  if (tid >= 16 && tid < 64) {
    int j = tid - 16;
    Xs[(129 + (j >> 4)) * XSTRIDE + (j & 15)] = 0.f;
  }

  // Preload this wave's W1/W2 slab as WMMA A-operands (registers, reused for
  // every point). A layout: lane holds row slab+(lane&15), K pair 2*hh+{0,1}.
  float w1a[33], w1b[33];
  {
    const float* wr = w1 + (slab + col) * 129;
#pragma unroll
    for (int kk = 0; kk < 33; ++kk) {
      int c0 = 4 * kk + 2 * hh;
      int c1 = c0 + 1;
      if (c0 > 128) c0 = 128;   // clamp: padded K rows of X are zero anyway
      if (c1 > 128) c1 = 128;
      w1a[kk] = wr[c0];
      w1b[kk] = wr[c1];
    }
  }
  float w2a[32], w2b[32];
  {
    const float* wr = w2 + (slab + col) * 128;
#pragma unroll
    for (int kk = 0; kk < 32; ++kk) {
      w2a[kk] = wr[4 * kk + 2 * hh];
      w2b[kk] = wr[4 * kk + 2 * hh + 1];
    }
  }
  __syncthreads();

  const int npts = BDIM * MDIM;
  for (int p = blockIdx.x; p < npts; p += gridDim.x) {
    const int b = p >> 13;        // p / 8192
    const int m = p & 8191;

    if (tid < 16) nbr[tid] = knn_idx[(size_t)p * KNN + tid];
    __syncthreads();

    // Gather neighbor features: half-wave (16 lanes) copies 64B chunks of one
    // neighbor's contiguous 512B feature row from featsT.
    {
      const int k  = tid >> 4;
      const int cb = tid & 15;
      const float* fr = featsT + ((size_t)b * NDIM + nbr[k]) * CDIM;
#pragma unroll
      for (int c = 0; c < CDIM; c += 16)
        Xs[(c + cb) * XSTRIDE + k] = fr[c + cb];
    }
    if (tid < 16) Xs[128 * XSTRIDE + tid] = knn_dist[(size_t)p * KNN + tid];
    __syncthreads();

    // ---- GEMM1: (16x132 slab of W1) x (132x16 X) via 33 f32 WMMA steps ----
    v8f acc1 = {};
#pragma unroll
    for (int kk = 0; kk < 33; ++kk) {
      const int r0 = 4 * kk + 2 * hh;
      v2f a; a[0] = w1a[kk];                 a[1] = w1b[kk];
      v2f bb; bb[0] = Xs[r0 * XSTRIDE + col]; bb[1] = Xs[(r0 + 1) * XSTRIDE + col];
      acc1 = __builtin_amdgcn_wmma_f32_16x16x4_f32(false, a, false, bb,
                                                   (short)0, acc1, false, false);
    }
    // Epilogue 1: BN+bias+lrelu, stage y1 into LDS for GEMM2 + s1
#pragma unroll
    for (int r = 0; r < 8; ++r) {
      const int c = slab + r + 8 * hh;
      float v = acc1[r] * sc1s[c] + sh1s[c];
      v = v > 0.f ? v : 0.2f * v;
      Ys[c * XSTRIDE + col] = v;
    }
    __syncthreads();

    if (tid < 128) {                       // s1 = sum over K
      float s = 0.f;
#pragma unroll
      for (int k = 0; k < KNN; ++k) s += Ys[tid * XSTRIDE + k];
      Ss[tid] = s;
    }

    // ---- GEMM2: (16x128 slab of W2) x (128x16 y1) via 32 f32 WMMA steps ---
    v8f acc2 = {};
#pragma unroll
    for (int kk = 0; kk < 32; ++kk) {
      const int r0 = 4 * kk + 2 * hh;
      v2f a; a[0] = w2a[kk];                 a[1] = w2b[kk];
      v2f bb; bb[0] = Ys[r0 * XSTRIDE + col]; bb[1] = Ys[(r0 + 1) * XSTRIDE + col];
      acc2 = __builtin_amdgcn_wmma_f32_16x16x4_f32(false, a, false, bb,
                                                   (short)0, acc2, false, false);
    }
    __syncthreads();
    // Epilogue 2 -> reuse Xs rows 0..127 as y2 staging
#pragma unroll
    for (int r = 0; r < 8; ++r) {
      const int c = slab + r + 8 * hh;
      float v = acc2[r] * sc2s[c] + sh2s[c];
      v = v > 0.f ? v : 0.2f * v;
      Xs[c * XSTRIDE + col] = v;
    }
    __syncthreads();
    if (tid < 128) {                       // s2 = sum over K
      float s = 0.f;
#pragma unroll
      for (int k = 0; k < KNN; ++k) s += Xs[tid * XSTRIDE + k];
      Ss[128 + tid] = s;
    }
    __syncthreads();

    // ---- conv3: out[c] = lrelu((W3[c,:]·[s1;s2])*sc3 + sh3), coalesced W3T
    if (tid < 128) {
      const int c = tid;
      float acc = 0.f;
#pragma unroll 8
      for (int i = 0; i < 256; ++i)
        acc = fmaf(w3t[i * CDIM + c], Ss[i], acc);
      float v = acc * sc3s[c] + sh3s[c];
      v = v > 0.f ? v : 0.2f * v;
      out[((size_t)b * CDIM + c) * MDIM + m] = v;
    }
    __syncthreads();   // protect nbr/Xs/Ss before next point
  }
}

// ---------------------------------------------------------------------------
extern "C" void kernel_launch(void* const* d_in, const int* in_sizes, int n_in,
                              void* d_out, int out_size, void* d_ws, size_t ws_size,
                              hipStream_t stream) {
  const float* atom_feats = (const float*)d_in[0];
  const float* atom_xyz   = (const float*)d_in[1];
  const float* surf_xyz   = (const float*)d_in[2];
  const float* w1  = (const float*)d_in[3];
  const float* b1  = (const float*)d_in[4];
  const float* g1  = (const float*)d_in[5];
  const float* be1 = (const float*)d_in[6];
  const float* mu1 = (const float*)d_in[7];
  const float* va1 = (const float*)d_in[8];
  const float* w2  = (const float*)d_in[9];
  const float* b2  = (const float*)d_in[10];
  const float* g2  = (const float*)d_in[11];
  const float* be2 = (const float*)d_in[12];
  const float* mu2 = (const float*)d_in[13];
  const float* va2 = (const float*)d_in[14];
  const float* w3  = (const float*)d_in[15];
  const float* b3  = (const float*)d_in[16];
  const float* g3  = (const float*)d_in[17];
  const float* be3 = (const float*)d_in[18];
  const float* mu3 = (const float*)d_in[19];
  const float* va3 = (const float*)d_in[20];
  float* out = (float*)d_out;

  // workspace partition (floats): featsT | w3t | knn_dist | knn_idx  (~12.7MB)
  float* featsT = (float*)d_ws;
  float* w3t    = featsT + (size_t)BDIM * NDIM * CDIM;
  float* kdist  = w3t + 256 * CDIM;
  int*   kidx   = (int*)(kdist + (size_t)BDIM * MDIM * KNN);

  dim3 tb(32, 8);
  dim3 tg(NDIM / 32, CDIM / 32, BDIM);
  feats_transpose_kernel<<<tg, tb, 0, stream>>>(atom_feats, featsT);
  w3_transpose_kernel<<<(256 * CDIM) / 256, 256, 0, stream>>>(w3, w3t);
  knn_kernel<<<(BDIM * MDIM) / 256, 256, 0, stream>>>(atom_xyz, surf_xyz,
                                                      kidx, kdist);
  fused_kernel<<<4096, 256, 0, stream>>>(featsT, kidx, kdist,
                                         w1, b1, g1, be1, mu1, va1,
                                         w2, b2, g2, be2, mu2, va2,
                                         w3t, b3, g3, be3, mu3, va3,
                                         out);
}